// TwoSimplicialAttention_80453327389223
// MI455X (gfx1250) — compile-verified
//
#include <hip/hip_runtime.h>
#include <hip/hip_bf16.h>
#include <math.h>

// ---------------------------------------------------------------------------
// TwoSimplicialAttention for MI455X (gfx1250, wave32, WMMA)
//
// Key structural fact: the W1=W2=32 windows are the LAST 32 sequence
// positions; the mask fully masks every query s < S-32, so P==0 and
// out[s] = LayerNorm(x[s]) for those rows.  Only the last 32 rows per batch
// run the projections / trilinear attention / W_O matmul.
//
// Roofline: ~54 MB of unavoidable traffic (x in, out, weights) vs <1.5 GFLOP
// of compute -> memory bound (~2.3 us @ 23.3 TB/s).  Since compute is free,
// the GEMMs use the full-precision f32 matrix op V_WMMA_F32_16X16X4_F32
// (matches the f32 reference exactly in data type; no bf16 rounding).
// ---------------------------------------------------------------------------

typedef __attribute__((ext_vector_type(2))) float v2f;
typedef __attribute__((ext_vector_type(8))) float v8f;

constexpr int B_   = 2;
constexpr int S_   = 2048;
constexpr int IN_  = 1024;
constexpr int H_   = 16;
constexpr int D_   = 64;
constexpr int OUTC = 1024;   // H*D
constexpr int W1_  = 32;
constexpr int SW_  = S_ - W1_;   // 2016 = window start

// workspace layout (float offsets); total 393216 floats = 1.5 MiB
constexpr size_t QN   = 0;        // [B][H][32][64] normalized Q
constexpr size_t KN   = 65536;    // [B][H][32][64] normalized K
constexpr size_t KPN  = 131072;   // [B][H][32][64] normalized K'
constexpr size_t VW   = 196608;   // [B][H][32][64] V window
constexpr size_t ZOFF = 262144;   // [B][32][1024]  Z
constexpr size_t OPRE = 327680;   // [B][32][1024]  Z@W_O + x  (pre-LN)

// ---------------------------------------------------------------------------
// V_WMMA_F32_16X16X4_F32 fragment layouts (ISA 7.12.2, wave32):
//   A 16x4 (f32):  lane l holds row m=l&15; koff=(l<16)?0:2;
//                  vgpr0 -> K=koff, vgpr1 -> K=koff+1
//   B 4x16 (f32):  lane l holds col n=l&15; koff=(l<16)?0:2;
//                  vgpr0 -> K=koff, vgpr1 -> K=koff+1
//   C/D 16x16:     lane l col n=l&15; vgpr r -> row m = ((l<16)?0:8)+r
// ---------------------------------------------------------------------------

__device__ __forceinline__ float wred_max(float v) {
#pragma unroll
  for (int off = 16; off > 0; off >>= 1) v = fmaxf(v, __shfl_xor(v, off, 32));
  return v;
}
__device__ __forceinline__ float wred_sum(float v) {
#pragma unroll
  for (int off = 16; off > 0; off >>= 1) v += __shfl_xor(v, off, 32);
  return v;
}

// ---------------------------------------------------------------------------
// Kernel 1: projections  P = x_win(32x1024) @ W[:, head]  -> 32x64 block,
// then per-row L2 norm (Q,K,Kp) or passthrough (V).  One block per
// (batch, weight, head): 2*4*16 = 128 blocks, 8 waves, 1 WMMA tile each.
// ---------------------------------------------------------------------------
__global__ __launch_bounds__(256) void proj_kernel(
    const float* __restrict__ x,
    const float* __restrict__ Wq, const float* __restrict__ Wk,
    const float* __restrict__ Wkp, const float* __restrict__ Wv,
    float* __restrict__ ws) {
  __shared__ float res[32][64];
  __shared__ float sinv[32];

  const int bi = blockIdx.x;
  const int b  = bi >> 6;          // /(4*H)
  const int w  = (bi >> 4) & 3;    // which weight
  const int h  = bi & 15;          // head

  const float* Wsel = (w == 0) ? Wq : (w == 1) ? Wk : (w == 2) ? Wkp : Wv;
  const size_t dsto = (w == 0) ? QN : (w == 1) ? KN : (w == 2) ? KPN : VW;
  float* dst = ws + dsto + ((size_t)(b * H_ + h)) * W1_ * D_;
  const bool do_norm = (w < 3);

  const float* xw = x + ((size_t)b * S_ + SW_) * IN_;  // 32 x 1024 window

  const int tid  = threadIdx.x;
  const int wave = tid >> 5;
  const int lane = tid & 31;
  const int mt = wave >> 2;            // 0..1 : 16-row tile
  const int nt = wave & 3;             // 0..3 : 16-col tile within head
  const int m0 = mt * 16;
  const int ml = lane & 15;
  const int am = m0 + ml;              // A row for this lane
  const int koff = (lane < 16) ? 0 : 2;
  const int bn = h * 64 + nt * 16 + ml;  // B (weight) column

  const float* arow = xw + (size_t)am * IN_ + koff;
  const float* bcol = Wsel + (size_t)koff * OUTC + bn;

  v8f acc = {};
#pragma unroll 8
  for (int kk = 0; kk < IN_; kk += 4) {
    v2f afrag, bfrag;
    afrag[0] = arow[kk];
    afrag[1] = arow[kk + 1];
    bfrag[0] = bcol[(size_t)kk * OUTC];
    bfrag[1] = bcol[(size_t)kk * OUTC + OUTC];
    if ((kk & 31) == 0 && kk + 32 < IN_)
      __builtin_prefetch(&bcol[(size_t)(kk + 32) * OUTC], 0, 1);
    acc = __builtin_amdgcn_wmma_f32_16x16x4_f32(
        false, afrag, false, bfrag, (short)0, acc, false, false);
  }

  // scatter D tile into LDS result block
  const int cm = m0 + ((lane >> 4) ? 8 : 0);
  const int cc = nt * 16 + ml;
#pragma unroll
  for (int r = 0; r < 8; ++r) res[cm + r][cc] = acc[r];
  __syncthreads();

  // per-row L2 norm (over D=64) for Q/K/Kp
  if (tid < 32) {
    float ss = 0.f;
#pragma unroll
    for (int d = 0; d < 64; ++d) ss += res[tid][d] * res[tid][d];
    sinv[tid] = do_norm ? (1.f / (sqrtf(ss) + 1e-7f)) : 1.f;
  }
  __syncthreads();

#pragma unroll
  for (int i = 0; i < 8; ++i) {
    const int e = tid + 256 * i;      // 2048 elements
    const int r = e >> 6, d = e & 63;
    dst[(size_t)r * D_ + d] = res[r][d] * sinv[r];
  }
}

// ---------------------------------------------------------------------------
// Kernel 2: trilinear attention on the 32-row window, per (b,h).
// A[s,j,k] = sum_d Q[s,d] K[j,d] Kp[k,d] / 8, causal mask j,k <= s,
// joint softmax over 32*32, Z[s,d] = sum_j V[j,d] * sum_k P[j,k] V[k,d].
// 32 blocks x 256 threads (8 waves); wave handles s = wave + 8*si.
// ---------------------------------------------------------------------------
__global__ __launch_bounds__(256) void attn_kernel(float* __restrict__ ws) {
  __shared__ float Ks[32][64];
  __shared__ float Kps[32][64];
  __shared__ float Vs[32][64];
  __shared__ float Ps[8][32][32];   // per-wave P tile

  const int b = blockIdx.x >> 4;
  const int h = blockIdx.x & 15;
  const size_t base = ((size_t)(b * H_ + h)) * W1_ * D_;
  const float* wsQ = ws + QN + base;
  const float* wsK = ws + KN + base;
  const float* wsKp = ws + KPN + base;
  const float* wsV = ws + VW + base;
  float* zout = ws + ZOFF + (size_t)b * W1_ * OUTC;

  const int tid = threadIdx.x;
  for (int e = tid; e < 32 * 64; e += 256) {
    (&Ks[0][0])[e]  = wsK[e];
    (&Kps[0][0])[e] = wsKp[e];
    (&Vs[0][0])[e]  = wsV[e];
  }
  __syncthreads();

  const int wave = tid >> 5;
  const int j    = tid & 31;     // lane

  for (int si = 0; si < 4; ++si) {
    const int s = wave + 8 * si;
    // ---- phase 1: scores + softmax (lanes = j) ----
    const float* qrow = wsQ + (size_t)s * D_;
    float qk[64];
#pragma unroll
    for (int d = 0; d < 64; ++d) qk[d] = qrow[d] * Ks[j][d];

    float lmax = -3.0e38f;
    for (int k = 0; k < 32; ++k) {
      float a = 0.f;
#pragma unroll
      for (int d = 0; d < 64; ++d) a += qk[d] * Kps[k][d];
      a *= 0.125f;                       // 1/sqrt(D)
      if ((j > s) || (k > s)) a = -3.0e38f;
      Ps[wave][j][k] = a;
      lmax = fmaxf(lmax, a);
    }
    lmax = wred_max(lmax);
    float lsum = 0.f;
    for (int k = 0; k < 32; ++k) {
      const float e = __expf(Ps[wave][j][k] - lmax);
      Ps[wave][j][k] = e;
      lsum += e;
    }
    const float inv = 1.f / wred_sum(lsum);   // (0,0) always unmasked
    __syncthreads();

    // ---- phase 2: Z (lanes = d pair) ----
    const int d0 = j, d1 = j + 32;
    float z0 = 0.f, z1 = 0.f;
    for (int jj = 0; jj < 32; ++jj) {
      float t0 = 0.f, t1 = 0.f;
      for (int k = 0; k < 32; ++k) {
        const float p = Ps[wave][jj][k];
        t0 += p * Vs[k][d0];
        t1 += p * Vs[k][d1];
      }
      z0 += Vs[jj][d0] * t0;
      z1 += Vs[jj][d1] * t1;
    }
    zout[(size_t)s * OUTC + h * D_ + d0] = z0 * inv;
    zout[(size_t)s * OUTC + h * D_ + d1] = z1 * inv;
    __syncthreads();
  }
}

// ---------------------------------------------------------------------------
// Kernel 3: out_pre = Z(32x1024) @ W_O(1024x1024) + x_win, f32 WMMA.
// 16 blocks = 2 batches x 8 column groups (128 cols); 8 waves, each owns one
// 16-col tile across both 16-row tiles (2 accumulators).
// ---------------------------------------------------------------------------
__global__ __launch_bounds__(256) void ogemm_kernel(
    const float* __restrict__ x, const float* __restrict__ Wo,
    float* __restrict__ ws) {
  const int bi = blockIdx.x;
  const int b  = bi >> 3;
  const int cg = bi & 7;

  const int tid  = threadIdx.x;
  const int wave = tid >> 5;
  const int lane = tid & 31;
  const int col0 = cg * 128 + wave * 16;
  const int ml   = lane & 15;
  const int koff = (lane < 16) ? 0 : 2;
  const int bn   = col0 + ml;

  const float* zb = ws + ZOFF + (size_t)b * W1_ * OUTC;
  float*       op = ws + OPRE + (size_t)b * W1_ * OUTC;
  const float* xw = x + ((size_t)b * S_ + SW_) * IN_;

  const float* a0row = zb + (size_t)ml * OUTC + koff;
  const float* a1row = zb + (size_t)(ml + 16) * OUTC + koff;
  const float* bcol  = Wo + (size_t)koff * OUTC + bn;

  v8f acc0 = {}, acc1 = {};
#pragma unroll 8
  for (int kk = 0; kk < OUTC; kk += 4) {
    v2f a0, a1, bfrag;
    a0[0] = a0row[kk];
    a0[1] = a0row[kk + 1];
    a1[0] = a1row[kk];
    a1[1] = a1row[kk + 1];
    bfrag[0] = bcol[(size_t)kk * OUTC];
    bfrag[1] = bcol[(size_t)kk * OUTC + OUTC];
    if ((kk & 31) == 0 && kk + 32 < OUTC)
      __builtin_prefetch(&bcol[(size_t)(kk + 32) * OUTC], 0, 1);
    acc0 = __builtin_amdgcn_wmma_f32_16x16x4_f32(
        false, a0, false, bfrag, (short)0, acc0, false, false);
    acc1 = __builtin_amdgcn_wmma_f32_16x16x4_f32(
        false, a1, false, bfrag, (short)0, acc1, false, false);
  }

  const int cm = (lane >> 4) ? 8 : 0;
  const int cc = col0 + ml;
#pragma unroll
  for (int r = 0; r < 8; ++r) {
    const int m0 = cm + r;
    op[(size_t)m0 * OUTC + cc] = acc0[r] + xw[(size_t)m0 * OUTC + cc];
    const int m1 = m0 + 16;
    op[(size_t)m1 * OUTC + cc] = acc1[r] + xw[(size_t)m1 * OUTC + cc];
  }
}

// ---------------------------------------------------------------------------
// Kernel 4: streaming LayerNorm over all B*S rows (bandwidth-critical).
// One wave per 1024-element row; float4 vector loads/stores; shfl reductions.
// Rows s >= 2016 read the attention output from workspace, others read x
// (residual of zero-Z rows is just x).
// ---------------------------------------------------------------------------
__global__ __launch_bounds__(256) void ln_kernel(
    const float* __restrict__ x, const float* __restrict__ gamma,
    const float* __restrict__ beta, const float* __restrict__ ws,
    float* __restrict__ out) {
  const int tid  = threadIdx.x;
  const int wave = tid >> 5;
  const int lane = tid & 31;
  const int row  = blockIdx.x * 8 + wave;     // 0..4095
  const int b = row >> 11;
  const int s = row & 2047;

  const float* src = (s >= SW_)
      ? (ws + OPRE + ((size_t)(b * W1_ + (s - SW_))) * OUTC)
      : (x + (size_t)row * OUTC);
  const float4* s4 = (const float4*)src;
  const float4* g4 = (const float4*)gamma;
  const float4* be4 = (const float4*)beta;
  float4* o4 = (float4*)(out + (size_t)row * OUTC);

  float4 v[8];
  float sum = 0.f;
#pragma unroll
  for (int i = 0; i < 8; ++i) {
    v[i] = s4[i * 32 + lane];
    sum += v[i].x + v[i].y + v[i].z + v[i].w;
  }
  const float mu = wred_sum(sum) * (1.f / 1024.f);

  float vs = 0.f;
#pragma unroll
  for (int i = 0; i < 8; ++i) {
    const float dx = v[i].x - mu, dy = v[i].y - mu;
    const float dz = v[i].z - mu, dw = v[i].w - mu;
    vs += dx * dx + dy * dy + dz * dz + dw * dw;
  }
  const float var = wred_sum(vs) * (1.f / 1024.f);
  const float rstd = rsqrtf(var + 1e-5f);

#pragma unroll
  for (int i = 0; i < 8; ++i) {
    const int c4 = i * 32 + lane;
    const float4 g = g4[c4];
    const float4 bb = be4[c4];
    float4 o;
    o.x = (v[i].x - mu) * rstd * g.x + bb.x;
    o.y = (v[i].y - mu) * rstd * g.y + bb.y;
    o.z = (v[i].z - mu) * rstd * g.z + bb.z;
    o.w = (v[i].w - mu) * rstd * g.w + bb.w;
    o4[c4] = o;
  }
}

// ---------------------------------------------------------------------------
extern "C" void kernel_launch(void* const* d_in, const int* in_sizes, int n_in,
                              void* d_out, int out_size, void* d_ws,
                              size_t ws_size, hipStream_t stream) {
  const float* x     = (const float*)d_in[0];
  const float* Wq    = (const float*)d_in[1];
  const float* Wk    = (const float*)d_in[2];
  const float* Wv    = (const float*)d_in[3];
  const float* Wkp   = (const float*)d_in[4];
  const float* Wo    = (const float*)d_in[5];
  const float* gamma = (const float*)d_in[6];
  const float* beta  = (const float*)d_in[7];
  float* out = (float*)d_out;
  float* ws  = (float*)d_ws;   // needs 1.5 MiB

  proj_kernel<<<B_ * 4 * H_, 256, 0, stream>>>(x, Wq, Wk, Wkp, Wv, ws);
  attn_kernel<<<B_ * H_, 256, 0, stream>>>(ws);
  ogemm_kernel<<<B_ * 8, 256, 0, stream>>>(x, Wo, ws);
  ln_kernel<<<(B_ * S_) / 8, 256, 0, stream>>>(x, gamma, beta, ws, out);
}